// LogicLoss_49907519979777
// MI455X (gfx1250) — compile-verified
//
#include <hip/hip_runtime.h>
#include <hip/hip_bf16.h>
#include <stdint.h>

#define BB 4
#define SS 2048
#define VV 8192
#define NS 64
#define RR (BB * SS)        // 8192 rows
#define NCH 64              // chunks per batch for DFA scan
#define CHLEN (SS / NCH)    // 32 tokens per chunk
#define TILE 4096           // half-row tile (16 KB) for TDM double buffering
#define RPB 8               // rows per block in main kernel
#define NTILES (RPB * 2)
#define NWAVES 8            // 256 threads / wave32

typedef unsigned int v4u __attribute__((ext_vector_type(4)));
typedef int v8i __attribute__((ext_vector_type(8)));
typedef int v4i __attribute__((ext_vector_type(4)));

// ---------------------------------------------------------------------------
// TDM: 1-D async copy of `nelem` f32 from global to LDS (D# built in SGPRs).
// Group0: count=1 | lds_addr | global_addr | type=2
// Group1: data_size=4B, tensor_dim0=nelem, tensor_dim1=1, tile_dim0=nelem,
//         tile_dim1=1, stride0=nelem. Groups 2/3 unused (zero) for <=2D.
// 6-arg builtin flavor: (v4u g0, v8i g1, v4i g2, v4i g3, v8i g4, i32 cpol).
// ---------------------------------------------------------------------------
__device__ __forceinline__ void tdm_load_1d(uint32_t lds_addr, const void* gptr,
                                            uint32_t nelem) {
  uint64_t ga = (uint64_t)(uintptr_t)gptr;
  v4u g0;
  g0.x = 1u;                                            // count=1, user mode
  g0.y = lds_addr;                                      // LDS byte address
  g0.z = (uint32_t)ga;                                  // global addr lo
  g0.w = (uint32_t)((ga >> 32) & 0x01FFFFFFu) | (2u << 30);  // addr hi | type=2
  v8i g1;
  g1[0] = (int)(2u << 16);                              // data_size = 4 bytes
  g1[1] = (int)((nelem & 0xFFFFu) << 16);               // tensor_dim0[15:0]
  g1[2] = (int)((nelem >> 16) | (1u << 16));            // dim0[31:16] | dim1=1 lo
  g1[3] = (int)((nelem & 0xFFFFu) << 16);               // dim1 hi=0 | tile_dim0
  g1[4] = (int)1u;                                      // tile_dim1=1, tile_dim2=0
  g1[5] = (int)nelem;                                   // tensor_dim0_stride lo32
  g1[6] = 0;                                            // stride0 hi | stride1 lo
  g1[7] = 0;                                            // stride1 hi
  v4i z4 = {0, 0, 0, 0};
  v8i z8 = {0, 0, 0, 0, 0, 0, 0, 0};
  __builtin_amdgcn_tensor_load_to_lds(g0, g1, z4, z4, z8, 0);
}

// ---------------------------------------------------------------------------
// 1) reject[state][v] bitmask: bit = (state_types[transition[state][v]] == -1)
//    64 states x 8192 tokens = 64 KB, stays L2-resident.
// ---------------------------------------------------------------------------
__global__ __launch_bounds__(256) void k_reject(const int* __restrict__ trans,
                                                const int* __restrict__ stypes,
                                                unsigned* __restrict__ rej) {
  int st = blockIdx.x;
  int t = threadIdx.x;  // word index 0..255
  const int* rowp = trans + (size_t)st * VV + t * 32;
  unsigned w = 0u;
  for (int j = 0; j < 32; ++j)
    if (stypes[rowp[j]] == -1) w |= (1u << j);
  rej[st * (VV / 32) + t] = w;
}

// ---------------------------------------------------------------------------
// 2) token = argmax over V (first-max tie break), one block per row.
//    Wave32 shuffle butterfly + 8-entry cross-wave combine.
// ---------------------------------------------------------------------------
__global__ __launch_bounds__(256) void k_argmax(const float* __restrict__ inp,
                                                int* __restrict__ tok) {
  int row = blockIdx.x;
  int t = threadIdx.x;
  int wid = t >> 5, lane = t & 31;
  const float4* p = (const float4*)(inp + (size_t)row * VV);
  float bm = -INFINITY;
  int bi = 0x7fffffff;
  for (int i = 0; i < VV / 4 / 256; ++i) {
    int q = i * 256 + t;
    float4 v = p[q];
    int base = q * 4;
    if (v.x > bm) { bm = v.x; bi = base; }
    if (v.y > bm) { bm = v.y; bi = base + 1; }
    if (v.z > bm) { bm = v.z; bi = base + 2; }
    if (v.w > bm) { bm = v.w; bi = base + 3; }
  }
  for (int off = 16; off > 0; off >>= 1) {
    float ov = __shfl_xor(bm, off, 32);
    int oi = __shfl_xor(bi, off, 32);
    if (ov > bm || (ov == bm && oi < bi)) { bm = ov; bi = oi; }
  }
  __shared__ float sm[NWAVES];
  __shared__ int si[NWAVES];
  if (lane == 0) { sm[wid] = bm; si[wid] = bi; }
  __syncthreads();
  if (t == 0) {
    for (int k = 1; k < NWAVES; ++k) {
      float ov = sm[k]; int oi = si[k];
      if (ov > bm || (ov == bm && oi < bi)) { bm = ov; bi = oi; }
    }
    tok[row] = bi;
  }
}

// ---------------------------------------------------------------------------
// 3a) Per-(batch,chunk): composed transition function for all 64 start states.
//     64 threads, each a 32-deep dependent chain -- all 256 blocks in parallel.
// ---------------------------------------------------------------------------
__global__ __launch_bounds__(64) void k_chunkfn(const int* __restrict__ trans,
                                                const int* __restrict__ tok,
                                                unsigned char* __restrict__ cfn) {
  int c = blockIdx.x, b = blockIdx.y, s = threadIdx.x;
  const int* tokp = tok + b * SS + c * CHLEN;
  int cur = s;
  for (int i = 0; i < CHLEN; ++i) cur = trans[cur * VV + tokp[i]];
  cfn[(b * NCH + c) * NS + s] = (unsigned char)cur;
}

// 3b) Sequential composition over 64 chunk functions (tiny).
__global__ void k_scan(const unsigned char* __restrict__ cfn,
                       int* __restrict__ cstart) {
  int b = threadIdx.x;
  if (b < BB) {
    int s = 0;
    for (int c = 0; c < NCH; ++c) {
      cstart[b * NCH + c] = s;
      s = cfn[(b * NCH + c) * NS + s];
    }
  }
}

// 3c) Replay each chunk from its known start state -> state before each token.
__global__ __launch_bounds__(NCH) void k_states(const int* __restrict__ trans,
                                                const int* __restrict__ tok,
                                                const int* __restrict__ cstart,
                                                int* __restrict__ state_ids) {
  int b = blockIdx.x, c = threadIdx.x;
  int s = cstart[b * NCH + c];
  int base = b * SS + c * CHLEN;
  for (int i = 0; i < CHLEN; ++i) {
    state_ids[base + i] = s;
    s = trans[s * VV + tok[base + i]];
  }
}

// ---------------------------------------------------------------------------
// 4) Main pass over predictions: TDM double-buffered half-row tiles into LDS,
//    flash-style online softmax merge, reject-masked exp sum.
//    Reductions: wave32 shuffle butterfly + 8-entry LDS combine (4 barriers
//    per tile instead of ~18).
// ---------------------------------------------------------------------------
__global__ __launch_bounds__(256) void k_main(const float* __restrict__ pred,
                                              const int* __restrict__ targets,
                                              const int* __restrict__ state_ids,
                                              const unsigned* __restrict__ rej,
                                              float* __restrict__ rowA,
                                              float* __restrict__ rowW,
                                              float* __restrict__ rowI) {
  __shared__ __align__(16) float buf[2][TILE];
  __shared__ float swm[NWAVES];
  __shared__ float sws[NWAVES];
  __shared__ float swr[NWAVES];

  int t = threadIdx.x;
  int wid = t >> 5, lane = t & 31;  // wave32
  int row0 = blockIdx.x * RPB;
  uint32_t lds0 = (uint32_t)(uintptr_t)&buf[0][0];
  uint32_t lds1 = (uint32_t)(uintptr_t)&buf[1][0];

  if (wid == 0) tdm_load_1d(lds0, pred + (size_t)row0 * VV, TILE);

  float m = -INFINITY, sum = 0.f, rej_s = 0.f;
  const unsigned* rw = rej;

  for (int j = 0; j < NTILES; ++j) {
    int rrow = row0 + (j >> 1);
    if (wid == 0) {
      if (j + 1 < NTILES) {
        int jn = j + 1;
        const float* g = pred + (size_t)(row0 + (jn >> 1)) * VV + (jn & 1) * TILE;
        tdm_load_1d((jn & 1) ? lds1 : lds0, g, TILE);
        __builtin_amdgcn_s_wait_tensorcnt(1);  // tile j complete
      } else {
        __builtin_amdgcn_s_wait_tensorcnt(0);
      }
    }
    __syncthreads();  // tile j visible to all waves

    const float4* b4 = (const float4*)((j & 1) ? &buf[1][0] : &buf[0][0]);
    if ((j & 1) == 0) {
      m = -INFINITY; sum = 0.f; rej_s = 0.f;
      rw = rej + state_ids[rrow] * (VV / 32);
    }

    // ---- tile max: wave butterfly, then 8-wide combine ----
    float lm = -INFINITY;
    for (int i = 0; i < TILE / 4 / 256; ++i) {
      float4 v = b4[i * 256 + t];
      lm = fmaxf(lm, fmaxf(fmaxf(v.x, v.y), fmaxf(v.z, v.w)));
    }
    for (int off = 16; off > 0; off >>= 1)
      lm = fmaxf(lm, __shfl_xor(lm, off, 32));
    if (lane == 0) swm[wid] = lm;
    __syncthreads();
    float mt = swm[0];
    for (int k = 1; k < NWAVES; ++k) mt = fmaxf(mt, swm[k]);

    // ---- tile exp sums (plain + reject-masked) ----
    float ls = 0.f, lr = 0.f;
    int vbase = (j & 1) * TILE;
    for (int i = 0; i < TILE / 4 / 256; ++i) {
      int q = i * 256 + t;
      float4 v = b4[q];
      int v0 = vbase + q * 4;
      unsigned bits = rw[v0 >> 5] >> (v0 & 31);
      float e0 = __expf(v.x - mt), e1 = __expf(v.y - mt);
      float e2 = __expf(v.z - mt), e3 = __expf(v.w - mt);
      ls += (e0 + e1) + (e2 + e3);
      lr += ((bits & 1u) ? e0 : 0.f) + (((bits >> 1) & 1u) ? e1 : 0.f) +
            (((bits >> 2) & 1u) ? e2 : 0.f) + (((bits >> 3) & 1u) ? e3 : 0.f);
    }
    for (int off = 16; off > 0; off >>= 1) {
      ls += __shfl_xor(ls, off, 32);
      lr += __shfl_xor(lr, off, 32);
    }
    if (lane == 0) { sws[wid] = ls; swr[wid] = lr; }
    __syncthreads();
    float st_ = 0.f, rt_ = 0.f;
    for (int k = 0; k < NWAVES; ++k) { st_ += sws[k]; rt_ += swr[k]; }

    // ---- online merge into running row stats (uniform across threads) ----
    float nm = fmaxf(m, mt);
    float fa = __expf(m - nm), fb = __expf(mt - nm);
    sum = sum * fa + st_ * fb;
    rej_s = rej_s * fa + rt_ * fb;
    m = nm;

    if ((j & 1) == 1 && t == 0) {
      int tg = targets[rrow];
      float xt = pred[(size_t)rrow * VV + tg];           // single scalar load
      float ce = (m + __logf(sum)) - xt;                 // -log_softmax[tgt]
      unsigned wb = (rw[tg >> 5] >> (tg & 31)) & 1u;
      float w = wb ? 0.05f : 1.0f;
      rowA[rrow] = ce * w;
      rowW[rrow] = w;
      rowI[rrow] = rej_s / sum;
    }
    __syncthreads();  // buf[j&1] + swm/sws/swr reads done before reuse
  }
}

// ---------------------------------------------------------------------------
// 5) Final deterministic reduction -> scalar loss.
// ---------------------------------------------------------------------------
__global__ __launch_bounds__(256) void k_final(const float* __restrict__ rowA,
                                               const float* __restrict__ rowW,
                                               const float* __restrict__ rowI,
                                               float* __restrict__ out) {
  __shared__ float sa[256], sb[256], sc[256];
  int t = threadIdx.x;
  float a = 0.f, b = 0.f, c = 0.f;
  for (int i = t; i < RR; i += 256) { a += rowA[i]; b += rowW[i]; c += rowI[i]; }
  sa[t] = a; sb[t] = b; sc[t] = c;
  __syncthreads();
  for (int s = 128; s > 0; s >>= 1) {
    if (t < s) { sa[t] += sa[t + s]; sb[t] += sb[t + s]; sc[t] += sc[t + s]; }
    __syncthreads();
  }
  if (t == 0) {
    float wce = sa[0] / (sb[0] + 1e-6f);
    float im = sc[0] / (float)RR;
    float pen = -__logf(1.0f - im + 1e-6f);
    out[0] = 0.7f * wce + 0.3f * pen;
  }
}

// ---------------------------------------------------------------------------
extern "C" void kernel_launch(void* const* d_in, const int* in_sizes, int n_in,
                              void* d_out, int out_size, void* d_ws, size_t ws_size,
                              hipStream_t stream) {
  const float* predictions = (const float*)d_in[0];
  const int* targets       = (const int*)d_in[1];
  const float* inputs      = (const float*)d_in[2];
  const int* transition    = (const int*)d_in[3];
  const int* state_types   = (const int*)d_in[4];
  float* out = (float*)d_out;

  char* w = (char*)d_ws;
  int* tok           = (int*)w;            w += (size_t)RR * 4;
  int* state_ids     = (int*)w;            w += (size_t)RR * 4;
  unsigned* rej      = (unsigned*)w;       w += (size_t)NS * (VV / 32) * 4;
  unsigned char* cfn = (unsigned char*)w;  w += (size_t)BB * NCH * NS;
  int* cstart        = (int*)w;            w += (size_t)BB * NCH * 4;
  float* rowA        = (float*)w;          w += (size_t)RR * 4;
  float* rowW        = (float*)w;          w += (size_t)RR * 4;
  float* rowI        = (float*)w;          w += (size_t)RR * 4;

  k_reject<<<NS, 256, 0, stream>>>(transition, state_types, rej);
  k_argmax<<<RR, 256, 0, stream>>>(inputs, tok);
  k_chunkfn<<<dim3(NCH, BB), NS, 0, stream>>>(transition, tok, cfn);
  k_scan<<<1, 32, 0, stream>>>(cfn, cstart);
  k_states<<<BB, NCH, 0, stream>>>(transition, tok, cstart, state_ids);
  k_main<<<RR / RPB, 256, 0, stream>>>(predictions, targets, state_ids, rej,
                                       rowA, rowW, rowI);
  k_final<<<1, 256, 0, stream>>>(rowA, rowW, rowI, out);
}